// adder2d_21131239097113
// MI455X (gfx1250) — compile-verified
//
#include <hip/hip_runtime.h>
#include <stdint.h>

// Problem constants (x=(16,128,28,28) f32, W=(256,128,3,3) f32, stride=1, pad=1)
#define NIMG        16
#define CIN         128
#define HW          28
#define OCH         256
#define PIX_IMG     (HW * HW)          // 784
#define PTOT        (NIMG * PIX_IMG)   // 12544
#define DTOT        (CIN * 9)          // 1152
#define TILE        64                 // 64 pixels x 64 ochans per workgroup
#define CCHUNK      4                  // channels per d-chunk
#define DCHUNK      (CCHUNK * 9)       // 36 d-values per chunk
#define NCHUNKS     (CIN / CCHUNK)     // 32

typedef __attribute__((address_space(1))) int glb_i32;
typedef __attribute__((address_space(3))) int lds_i32;
typedef __attribute__((address_space(3))) float lds_f32;

#if defined(__has_builtin)
#if __has_builtin(__builtin_amdgcn_global_load_async_to_lds_b32)
#define HAVE_ASYNC_LDS_BUILTIN 1
#endif
#if __has_builtin(__builtin_amdgcn_s_wait_asynccnt)
#define HAVE_WAIT_ASYNC_BUILTIN 1
#endif
#endif

// One 4-byte per-lane gather: global -> LDS via the async DMA path (ASYNCcnt).
__device__ __forceinline__ void async_copy_f32(const float* g, float* l) {
#ifdef HAVE_ASYNC_LDS_BUILTIN
    __builtin_amdgcn_global_load_async_to_lds_b32((glb_i32*)g, (lds_i32*)l, 0, 0);
#else
    unsigned loff = (unsigned)(uintptr_t)(lds_f32*)l;  // LDS byte offset (AS3 ptr is 32-bit)
    asm volatile("global_load_async_to_lds_b32 %0, %1, off"
                 :: "v"(loff), "v"(g) : "memory");
#endif
}

__device__ __forceinline__ void wait_async_le18() {
#ifdef HAVE_WAIT_ASYNC_BUILTIN
    __builtin_amdgcn_s_wait_asynccnt(18);
#else
    asm volatile("s_wait_asynccnt 18" ::: "memory");
#endif
}
__device__ __forceinline__ void wait_async_0() {
#ifdef HAVE_WAIT_ASYNC_BUILTIN
    __builtin_amdgcn_s_wait_asynccnt(0);
#else
    asm volatile("s_wait_asynccnt 0" ::: "memory");
#endif
}

// Issue one d-chunk of async loads: 9 im2col-gather X loads + 9 W loads per thread.
// Uniform 18 ASYNCcnt increments per wave per chunk.
__device__ __forceinline__ void issue_chunk(const float* (&xsrc)[9], const int (&xstep)[9],
                                            const float*& wsrc,
                                            float* sXbuf, float* sWbuf,
                                            int lc, int lp, int wlc, int lo) {
#pragma unroll
    for (int tap = 0; tap < 9; ++tap) {
        async_copy_f32(xsrc[tap], &sXbuf[(lc * 9 + tap) * TILE + lp]);
        xsrc[tap] += xstep[tap];
    }
#pragma unroll
    for (int i = 0; i < 9; ++i) {
        async_copy_f32(wsrc + i, &sWbuf[(wlc * 9 + i) * TILE + lo]);
    }
    wsrc += DCHUNK;
}

__global__ void __launch_bounds__(256)
zero_ws_kernel(float* z) {
    if (threadIdx.x < 16) z[threadIdx.x] = 0.0f;
}

// out[n,o,ho,wo] = -sum_d |W[o,d] - im2col(x)[p,d]|
__global__ void __launch_bounds__(256)
adder2d_kernel(const float* __restrict__ x, const float* __restrict__ Wt,
               float* __restrict__ out, const float* __restrict__ zp) {
    // double-buffered LDS tiles, d-major so compute reads are ds_load_b128
    __shared__ float sX[2][DCHUNK * TILE];   // [buf][d][pixel]
    __shared__ float sW[2][DCHUNK * TILE];   // [buf][d][ochan]

    const int tid  = threadIdx.x;
    const int pblk = blockIdx.x;             // 196 pixel blocks of 64
    const int oblk = blockIdx.y;             // 4 ochan blocks of 64

    // ---- loader role: lane (lp = pixel 0..63, lc = channel-in-chunk 0..3) ----
    const int lp = tid & 63;
    const int lc = tid >> 6;
    const int pg   = pblk * TILE + lp;
    const int n    = pg / PIX_IMG;
    const int rem  = pg % PIX_IMG;
    const int ho   = rem / HW;
    const int wo   = rem % HW;

    const float* xsrc[9];
    int xstep[9];
#pragma unroll
    for (int kh = 0; kh < 3; ++kh) {
#pragma unroll
        for (int kw = 0; kw < 3; ++kw) {
            const int tap = kh * 3 + kw;
            const int h = ho + kh - 1;
            const int w = wo + kw - 1;
            const bool inb = ((unsigned)h < (unsigned)HW) && ((unsigned)w < (unsigned)HW);
            // chunk 0 address for channel c = lc; OOB taps read a zeroed scratch word
            xsrc[tap]  = inb ? (x + (size_t)n * (CIN * PIX_IMG) + (size_t)lc * PIX_IMG
                                  + h * HW + w)
                             : zp;
            xstep[tap] = inb ? (CCHUNK * PIX_IMG) : 0;   // advance 4 channels per chunk
        }
    }
    const int lo  = tid & 63;                // ochan lane 0..63
    const int wlc = tid >> 6;                // channel-in-chunk 0..3
    const float* wsrc = Wt + (size_t)(oblk * TILE + lo) * DTOT + wlc * 9;

    // ---- compute role: 4 pixels x 4 ochans per thread ----
    const int p0 = (tid & 15) * 4;
    const int o0 = (tid >> 4) * 4;

    float acc[4][4];
#pragma unroll
    for (int i = 0; i < 4; ++i)
#pragma unroll
        for (int j = 0; j < 4; ++j) acc[i][j] = 0.0f;

    // prefetch chunk 0 into buffer 0
    issue_chunk(xsrc, xstep, wsrc, sX[0], sW[0], lc, lp, wlc, lo);

#pragma unroll 1
    for (int cc = 0; cc < NCHUNKS; ++cc) {
        const int buf = cc & 1;
        if (cc + 1 < NCHUNKS) {
            issue_chunk(xsrc, xstep, wsrc, sX[buf ^ 1], sW[buf ^ 1], lc, lp, wlc, lo);
            wait_async_le18();   // oldest 18 (= chunk cc) have landed; next 18 in flight
        } else {
            wait_async_0();
        }
        __syncthreads();

        const float* sXbuf = sX[buf];
        const float* sWbuf = sW[buf];
#pragma unroll
        for (int dd = 0; dd < DCHUNK; ++dd) {
            const float4 xv = *(const float4*)(sXbuf + dd * TILE + p0);
            const float4 wv = *(const float4*)(sWbuf + dd * TILE + o0);
            acc[0][0] += __builtin_fabsf(wv.x - xv.x);
            acc[0][1] += __builtin_fabsf(wv.y - xv.x);
            acc[0][2] += __builtin_fabsf(wv.z - xv.x);
            acc[0][3] += __builtin_fabsf(wv.w - xv.x);
            acc[1][0] += __builtin_fabsf(wv.x - xv.y);
            acc[1][1] += __builtin_fabsf(wv.y - xv.y);
            acc[1][2] += __builtin_fabsf(wv.z - xv.y);
            acc[1][3] += __builtin_fabsf(wv.w - xv.y);
            acc[2][0] += __builtin_fabsf(wv.x - xv.z);
            acc[2][1] += __builtin_fabsf(wv.y - xv.z);
            acc[2][2] += __builtin_fabsf(wv.z - xv.z);
            acc[2][3] += __builtin_fabsf(wv.w - xv.z);
            acc[3][0] += __builtin_fabsf(wv.x - xv.w);
            acc[3][1] += __builtin_fabsf(wv.y - xv.w);
            acc[3][2] += __builtin_fabsf(wv.z - xv.w);
            acc[3][3] += __builtin_fabsf(wv.w - xv.w);
        }
        __syncthreads();   // everyone done reading buf before it is refilled
    }

    // ---- write out: out layout (N, O, Ho, Wo) ----
#pragma unroll
    for (int i = 0; i < 4; ++i) {
        const int p   = pblk * TILE + p0 + i;
        const int nn  = p / PIX_IMG;
        const int rr  = p % PIX_IMG;
        const size_t base = (size_t)nn * (OCH * PIX_IMG)
                          + (size_t)(oblk * TILE + o0) * PIX_IMG + rr;
#pragma unroll
        for (int j = 0; j < 4; ++j) {
            out[base + (size_t)j * PIX_IMG] = -acc[i][j];
        }
    }
}

extern "C" void kernel_launch(void* const* d_in, const int* in_sizes, int n_in,
                              void* d_out, int out_size, void* d_ws, size_t ws_size,
                              hipStream_t stream) {
    const float* x  = (const float*)d_in[0];   // (16,128,28,28) f32
    const float* Wt = (const float*)d_in[1];   // (256,128,3,3)  f32
    float* out = (float*)d_out;                // (16,256,28,28) f32
    float* zws = (float*)d_ws;                 // 64B zero pad source for halo lanes

    zero_ws_kernel<<<1, 64, 0, stream>>>(zws);

    dim3 grid(PTOT / TILE, OCH / TILE, 1);     // (196, 4)
    adder2d_kernel<<<grid, 256, 0, stream>>>(x, Wt, out, zws);
}